// MPLHGE_85547158602229
// MI455X (gfx1250) — compile-verified
//
#include <hip/hip_runtime.h>
#include <math.h>

typedef __attribute__((ext_vector_type(2))) float v2f;
typedef __attribute__((ext_vector_type(8))) float v8f;

#define D128 128

enum { EPI_NONE = 0, EPI_BIAS_RELU = 1, EPI_BIAS_TANH = 2 };

// ---------------------------------------------------------------------------
// WMMA GEMM: out[N,128] = f( A'[N,128] @ W[128,128] (+ bias) )
// A' = A (or A gathered by perm when GATHER). One wave computes a 16x128 strip
// using V_WMMA_F32_16X16X4_F32 (full fp32 precision, matches the reference).
// W (64KB) is staged in LDS K-quad-interleaved: Wq[k/4][n][4] so each B
// fragment (two consecutive K for one N column) is a single conflict-free
// ds_load_b64 with no register shuffles.
// ---------------------------------------------------------------------------
template<int EPI, bool GATHER>
__global__ __launch_bounds__(256)
void gemm128(const float* __restrict__ A, const float* __restrict__ W,
             const float* __restrict__ bias, float* __restrict__ out,
             int N, const int* __restrict__ perm)
{
    __shared__ float Wl[D128 * D128];          // 64 KB of the 320 KB WGP LDS
    // stage W K-quad-interleaved: Wl[(k>>2)*512 + n*4 + (k&3)] = W[k*128+n]
    for (int idx = threadIdx.x; idx < D128 * D128 / 4; idx += 256) {
        const int k  = idx >> 5;               // row of W (k index)
        const int n0 = (idx & 31) * 4;         // 4 consecutive n
        float4 w = *(const float4*)(W + k * D128 + n0);
        float* dst = Wl + (k >> 2) * 512 + n0 * 4 + (k & 3);
        dst[0]  = w.x;
        dst[4]  = w.y;
        dst[8]  = w.z;
        dst[12] = w.w;
    }
    __syncthreads();

    const int lane = threadIdx.x & 31;
    const int wid  = threadIdx.x >> 5;
    const int m0   = (blockIdx.x * 8 + wid) * 16;
    if (m0 >= N) return;

    const int mrow  = lane & 15;   // A row within tile / B,C column within tile
    const int hi    = lane >> 4;
    const int khalf = hi * 2;

    int arow = m0 + mrow;
    if (arow >= N) arow = N - 1;                         // clamped (stores masked)
    const long asrc = GATHER ? (long)perm[arow] : (long)arow;
    const float* __restrict__ Arow = A + asrc * D128;

    v8f acc[8];
#pragma unroll
    for (int t = 0; t < 8; ++t) acc[t] = 0.0f;

    for (int kk = 0; kk < D128; kk += 4) {
        // A frag: lane holds K = kk+2*hi, kk+2*hi+1 of its row (contiguous b64)
        v2f a = *(const v2f*)(Arow + kk + khalf);
        const float* Wk = Wl + (kk >> 2) * 512 + khalf;  // + n*4 per column
#pragma unroll
        for (int nt = 0; nt < 8; ++nt) {
            // B frag: one aligned ds_load_b64, conflict-free across the wave
            v2f b = *(const v2f*)(Wk + (nt * 16 + mrow) * 4);
            acc[nt] = __builtin_amdgcn_wmma_f32_16x16x4_f32(
                false, a, false, b, (short)0, acc[nt], false, false);
        }
    }

#pragma unroll
    for (int nt = 0; nt < 8; ++nt) {
        const int col = nt * 16 + mrow;
        const float bv = (EPI != EPI_NONE) ? bias[col] : 0.0f;
#pragma unroll
        for (int r = 0; r < 8; ++r) {
            const int row = m0 + r + 8 * hi;              // C layout: VGPR r -> M=r / r+8
            if (row < N) {
                float v = acc[nt][r];
                if (EPI == EPI_BIAS_RELU)      v = fmaxf(v + bv, 0.0f);
                else if (EPI == EPI_BIAS_TANH) v = tanhf(v + bv);
                out[(long)row * D128 + col] = v;
            }
        }
    }
}

// ---------------------------------------------------------------------------
// Small utility kernels
// ---------------------------------------------------------------------------
__global__ void fill0(float* p, long n) {
    long i = (long)blockIdx.x * blockDim.x + threadIdx.x;
    if (i < n) p[i] = 0.0f;
}

__global__ void deg_accum(const int* __restrict__ dst, long E, float* __restrict__ deg) {
    long e = (long)blockIdx.x * blockDim.x + threadIdx.x;
    if (e < E) atomicAdd(&deg[dst[e]], 1.0f);
}

__global__ void deg_to_dinv(float* d, long n) {
    long i = (long)blockIdx.x * blockDim.x + threadIdx.x;
    if (i < n) d[i] = rsqrtf(d[i] + 1.0f);               // +1 self loop
}

// z[i,:] = h[i,:] * dinv[i]^2 + bias   (self-loop message + bias, pre-scatter)
__global__ __launch_bounds__(256)
void prop_init(const float* __restrict__ h, const float* __restrict__ dinv,
               const float* __restrict__ bias, float* __restrict__ z, int N)
{
    long tid = (long)blockIdx.x * blockDim.x + threadIdx.x;
    long node = tid >> 5;
    int  c4   = (int)(tid & 31) * 4;
    if (node >= N) return;
    float di = dinv[node], s = di * di;
    float4 hv = *(const float4*)(h + node * D128 + c4);
    float4 b  = *(const float4*)(bias + c4);
    float4 o  = make_float4(fmaf(hv.x, s, b.x), fmaf(hv.y, s, b.y),
                            fmaf(hv.z, s, b.z), fmaf(hv.w, s, b.w));
    *(float4*)(z + node * D128 + c4) = o;
}

// z[dst,:] += h[src,:] * dinv[src]*dinv[dst].  Wave-per-edge: 32 lanes x float4
// = one coalesced 512B row; 4 global_atomic_add_f32 per lane.
__global__ __launch_bounds__(256)
void prop_scatter(const float* __restrict__ h, const int* __restrict__ src,
                  const int* __restrict__ dst, const float* __restrict__ dinv,
                  float* __restrict__ z, long E)
{
    long tid = (long)blockIdx.x * blockDim.x + threadIdx.x;
    long e = tid >> 5;
    int  c4 = (int)(tid & 31) * 4;
    if (e >= E) return;
    int s = src[e], d = dst[e];
    float nrm = dinv[s] * dinv[d];
    float4 hv = *(const float4*)(h + (long)s * D128 + c4);
    float* zp = z + (long)d * D128 + c4;
    atomicAdd(zp + 0, hv.x * nrm);
    atomicAdd(zp + 1, hv.y * nrm);
    atomicAdd(zp + 2, hv.z * nrm);
    atomicAdd(zp + 3, hv.w * nrm);
}

__global__ void relu4(float* __restrict__ z, long n4) {
    long i = (long)blockIdx.x * blockDim.x + threadIdx.x;
    if (i >= n4) return;
    float4 v = *(float4*)(z + i * 4);
    v.x = fmaxf(v.x, 0.0f); v.y = fmaxf(v.y, 0.0f);
    v.z = fmaxf(v.z, 0.0f); v.w = fmaxf(v.w, 0.0f);
    *(float4*)(z + i * 4) = v;
}

// acc += sum_rows( T[row,:] . q )  (T = tanh(z@W_sem+b_sem) from GEMM epilogue)
__global__ __launch_bounds__(256)
void rowdot_q(const float* __restrict__ T, const float* __restrict__ q,
              int N, float* __restrict__ acc)
{
    __shared__ float part[8];
    long tid = (long)blockIdx.x * blockDim.x + threadIdx.x;
    long row = tid >> 5;
    int  lane = (int)(tid & 31);
    float s = 0.0f;
    if (row < N) {
        float4 t  = *(const float4*)(T + row * D128 + lane * 4);
        float4 qv = *(const float4*)(q + lane * 4);
        s = t.x * qv.x + t.y * qv.y + t.z * qv.z + t.w * qv.w;
    }
    for (int o = 16; o > 0; o >>= 1) s += __shfl_down(s, o, 32);
    if (lane == 0) part[threadIdx.x >> 5] = s;
    __syncthreads();
    if (threadIdx.x == 0) {
        float t = 0.0f;
        for (int i = 0; i < 8; ++i) t += part[i];
        atomicAdd(acc, t);
    }
}

__global__ void softmax3(const float* __restrict__ sAcc, int N, float* __restrict__ w) {
    float m0 = sAcc[0] / (float)N, m1 = sAcc[1] / (float)N, m2 = sAcc[2] / (float)N;
    float mx = fmaxf(m0, fmaxf(m1, m2));
    float e0 = expf(m0 - mx), e1 = expf(m1 - mx), e2 = expf(m2 - mx);
    float inv = 1.0f / (e0 + e1 + e2);
    w[0] = e0 * inv; w[1] = e1 * inv; w[2] = e2 * inv;
}

// colsum[c] += sum_{b in chunk} sum_t w[t]*z_t[b,c]   (= column sums of att_pos[:B])
__global__ __launch_bounds__(128)
void wcolsum(const float* __restrict__ z0, const float* __restrict__ z1,
             const float* __restrict__ z2, const float* __restrict__ w,
             int B, float* __restrict__ colsum)
{
    const int R = 256;
    int c = threadIdx.x;
    float w0 = w[0], w1 = w[1], w2 = w[2];
    long r0 = (long)blockIdx.x * R;
    float s = 0.0f;
    for (int i = 0; i < R; ++i) {
        long b = r0 + i;
        if (b < B) {
            long o = b * D128 + c;
            s += w0 * z0[o] + w1 * z1[o] + w2 * z2[o];
        }
    }
    atomicAdd(&colsum[c], s);
}

// summary = 0.5*(sv + colsum/B) -> out;  c = W_disc @ summary
__global__ __launch_bounds__(128)
void summary_c(const float* __restrict__ colsum, const float* __restrict__ sv, int B,
               const float* __restrict__ Wdisc, float* __restrict__ cvec,
               float* __restrict__ out_summary)
{
    __shared__ float sm[D128];
    int d = threadIdx.x;
    float s = 0.5f * (sv[d] + colsum[d] / (float)B);
    sm[d] = s;
    out_summary[d] = s;
    __syncthreads();
    float ci = 0.0f;
    for (int k = 0; k < D128; ++k) ci = fmaf(Wdisc[d * D128 + k], sm[k], ci);
    cvec[d] = ci;
}

// out[b] = (sum_t w[t]*z_t[b,:]) . c + b_disc   (wave-per-row)
__global__ __launch_bounds__(256)
void score_out(const float* __restrict__ z0, const float* __restrict__ z1,
               const float* __restrict__ z2, const float* __restrict__ w,
               const float* __restrict__ c, const float* __restrict__ b_disc,
               float* __restrict__ outp, int B)
{
    long tid = (long)blockIdx.x * blockDim.x + threadIdx.x;
    long b = tid >> 5;
    int  lane = (int)(tid & 31);
    if (b >= B) return;
    float w0 = w[0], w1 = w[1], w2 = w[2];
    long off = b * D128 + lane * 4;
    float4 a0 = *(const float4*)(z0 + off);
    float4 a1 = *(const float4*)(z1 + off);
    float4 a2 = *(const float4*)(z2 + off);
    float4 cv = *(const float4*)(c + lane * 4);
    float s = (w0 * a0.x + w1 * a1.x + w2 * a2.x) * cv.x
            + (w0 * a0.y + w1 * a1.y + w2 * a2.y) * cv.y
            + (w0 * a0.z + w1 * a1.z + w2 * a2.z) * cv.z
            + (w0 * a0.w + w1 * a1.w + w2 * a2.w) * cv.w;
    for (int o = 16; o > 0; o >>= 1) s += __shfl_down(s, o, 32);
    if (lane == 0) outp[b] = s + b_disc[0];
}

// ---------------------------------------------------------------------------
extern "C" void kernel_launch(void* const* d_in, const int* in_sizes, int n_in,
                              void* d_out, int out_size, void* d_ws, size_t ws_size,
                              hipStream_t stream)
{
    const float* x      = (const float*)d_in[0];
    const int*   ei     = (const int*)d_in[1];   // [3,2,E]
    const int*   perm   = (const int*)d_in[2];
    const float* sv     = (const float*)d_in[3];
    // d_in[4] = Batch_size (recovered from out_size instead)
    const float* W_lin  = (const float*)d_in[5];
    const float* b_lin  = (const float*)d_in[6];
    const float* W1     = (const float*)d_in[7];
    const float* b1     = (const float*)d_in[8];
    const float* W2     = (const float*)d_in[9];
    const float* b2     = (const float*)d_in[10];
    const float* W_sem  = (const float*)d_in[11];
    const float* b_sem  = (const float*)d_in[12];
    const float* q_sem  = (const float*)d_in[13];
    const float* W_disc = (const float*)d_in[14];
    const float* b_disc = (const float*)d_in[15];
    float* out = (float*)d_out;

    const int  N = in_sizes[0] / D128;
    const long E = (long)in_sizes[1] / 6;          // 3 edge types x {src,dst}
    const int  B = (out_size - D128) / 2;          // out = [2B] ++ [128]

    // workspace layout (fp32)
    float* ws     = (float*)d_ws;
    float* dinv   = ws;                            // 3*N
    float* x0     = dinv + 3L * N;                 // N*128
    float* h      = x0   + (long)N * D128;         // N*128
    float* z[3]   = { h + (long)N * D128, h + 2L * N * D128, h + 3L * N * D128 };
    float* sAcc   = h + 4L * N * D128;             // 3 attention-score accumulators
    float* wtsP   = sAcc + 3;                      // 3
    float* wtsN   = wtsP + 3;                      // 3
    float* colsum = wtsN + 3;                      // 128
    float* cvec   = colsum + D128;                 // 128

    const int TB = 256;
    auto nb = [](long n, int tb) { return (unsigned)((n + tb - 1) / tb); };
    const unsigned gemmBlocks = (unsigned)(((N + 15) / 16 + 7) / 8);
    const long N32 = (long)N * 32, E32 = E * 32;

    // --- degrees / dinv per edge type (dst-only; shared by pos & neg branches)
    fill0<<<nb(3L * N, TB), TB, 0, stream>>>(dinv, 3L * N);
    for (int t = 0; t < 3; ++t) {
        const int* dst_t = ei + (long)t * 2 * E + E;
        deg_accum<<<nb(E, TB), TB, 0, stream>>>(dst_t, E, dinv + (long)t * N);
    }
    deg_to_dinv<<<nb(3L * N, TB), TB, 0, stream>>>(dinv, 3L * N);
    fill0<<<1, 128, 0, stream>>>(colsum, 128);

    auto run_branch = [&](bool gather, float* wts) {
        fill0<<<1, 32, 0, stream>>>(sAcc, 3);
        if (gather)
            gemm128<EPI_BIAS_RELU, true ><<<gemmBlocks, 256, 0, stream>>>(x, W_lin, b_lin, x0, N, perm);
        else
            gemm128<EPI_BIAS_RELU, false><<<gemmBlocks, 256, 0, stream>>>(x, W_lin, b_lin, x0, N, perm);
        for (int t = 0; t < 3; ++t) {
            const int* src_t = ei + (long)t * 2 * E;
            const int* dst_t = src_t + E;
            const float* dv  = dinv + (long)t * N;
            // conv1: h = x0@W1 ; z_t = prop(h) + b1 ; relu
            gemm128<EPI_NONE, false><<<gemmBlocks, 256, 0, stream>>>(x0, W1, nullptr, h, N, nullptr);
            prop_init   <<<nb(N32, TB), TB, 0, stream>>>(h, dv, b1, z[t], N);
            prop_scatter<<<nb(E32, TB), TB, 0, stream>>>(h, src_t, dst_t, dv, z[t], E);
            relu4       <<<nb(N32, TB), TB, 0, stream>>>(z[t], N32);
            // conv2: h = z_t@W2 ; z_t = prop(h) + b2 ; relu
            gemm128<EPI_NONE, false><<<gemmBlocks, 256, 0, stream>>>(z[t], W2, nullptr, h, N, nullptr);
            prop_init   <<<nb(N32, TB), TB, 0, stream>>>(h, dv, b2, z[t], N);
            prop_scatter<<<nb(E32, TB), TB, 0, stream>>>(h, src_t, dst_t, dv, z[t], E);
            relu4       <<<nb(N32, TB), TB, 0, stream>>>(z[t], N32);
        }
        // semantic attention scalars: sAcc[t] = sum_n tanh(z_t@W_sem+b_sem).q
        for (int t = 0; t < 3; ++t) {
            gemm128<EPI_BIAS_TANH, false><<<gemmBlocks, 256, 0, stream>>>(z[t], W_sem, b_sem, h, N, nullptr);
            rowdot_q<<<nb(N32, TB), TB, 0, stream>>>(h, q_sem, N, sAcc + t);
        }
        softmax3<<<1, 1, 0, stream>>>(sAcc, N, wts);
    };

    // positive branch -> summary, c, sc_pos
    run_branch(false, wtsP);
    wcolsum  <<<nb(B, 256), 128, 0, stream>>>(z[0], z[1], z[2], wtsP, B, colsum);
    summary_c<<<1, 128, 0, stream>>>(colsum, sv, B, W_disc, cvec, out + 2L * B);
    score_out<<<nb((long)B * 32, TB), TB, 0, stream>>>(z[0], z[1], z[2], wtsP, cvec, b_disc, out, B);

    // negative branch (reuses x0/h/z buffers; c is already fixed) -> sc_neg
    run_branch(true, wtsN);
    score_out<<<nb((long)B * 32, TB), TB, 0, stream>>>(z[0], z[1], z[2], wtsN, cvec, b_disc, out + B, B);
}